// QuantumKernelEmbedding_65481071407817
// MI455X (gfx1250) — compile-verified
//
#include <hip/hip_runtime.h>
#include <math.h>

// Quantum kernel embedding: out[b,s,r] = | prod_w cos(0.5*(x[b,s,w] - refs[r,w])) |
// B=64, S=32768, R=4, W=4  -> 2.1M "points", each float4 in, float4 out.
//
// Strategy (MI455X, gfx1250, wave32):
//  - Memory bound: 67 MB @ 23.3 TB/s ~ 2.9us. Keep ALU under that.
//  - cos((x-y)/2) = cos(x/2)cos(y/2) + sin(x/2)sin(y/2): 8 trans/point.
//  - The 16 (r,w) combinations per point form a 16x8 * 8x16 f32 matmul per
//    16-point tile -> two chained V_WMMA_F32_16X16X4_F32 (K=8 as 2x K=4).
//  - Product over wires via shfl_xor within column groups of 4 lanes.
//  - LDS-transpose stage -> one coalesced 256B global_store_b128 per tile.

typedef float v8f __attribute__((ext_vector_type(8)));
typedef float v2f __attribute__((ext_vector_type(2)));

constexpr int kBlock = 256;           // 8 wave32s per block
constexpr int kWavesPerBlock = kBlock / 32;

__global__ __launch_bounds__(kBlock)
void qke_wmma_kernel(const float* __restrict__ x,
                     const float* __restrict__ refs,
                     float* __restrict__ out,
                     int numTiles, int totalPoints)
{
  __shared__ __align__(16) float stage[kWavesPerBlock][64];

  const int lane  = threadIdx.x & 31;
  const int wv    = threadIdx.x >> 5;
  const int gwave = blockIdx.x * kWavesPerBlock + wv;
  const int nWaves = gridDim.x * kWavesPerBlock;

  const int n    = lane & 15;   // D column this lane owns: n = 4*r + w
  const int r    = n >> 2;
  const int w    = n & 3;
  const int half = lane >> 4;   // which pair of K rows this lane supplies

  // ---- B matrices (constant over all tiles) -------------------------------
  // B1 carries cos/sin(ref/2) for wires 0,1 ; B2 for wires 2,3.
  // B layout (4x16 f32): lane = n + 16*(k>>1), vgpr = k&1.
  const float ra = 0.5f * refs[n];          // refs[r][w] flat = 4r+w = n
  const float cr = __cosf(ra);
  const float sr = __sinf(ra);
  v2f b1 = {0.0f, 0.0f};
  v2f b2 = {0.0f, 0.0f};
  if (w == half)     { b1.x = cr; b1.y = sr; }   // rows (c,s) of wire `half`
  if (w == half + 2) { b2.x = cr; b2.y = sr; }   // rows (c,s) of wire `half+2`

  const float4* __restrict__ x4   = reinterpret_cast<const float4*>(x);
  float4*       __restrict__ out4 = reinterpret_cast<float4*>(out);

  // ---- main tiled loop: one 16-point tile per wave per iteration ----------
  for (int t = gwave; t < numTiles; t += nWaves) {
    const int pbase = t << 4;                 // first point of tile
    const float4 xv = x4[pbase + n];          // point n's 4 wires (coalesced)

    // A layout (16x4 f32): lanes 0-15 -> K0,K1 (wire `half`=0),
    //                      lanes 16-31 -> K2,K3 (wire `half`=1); A2 = wires 2/3.
    const float h0 = 0.5f * (half ? xv.y : xv.x);   // wire = half
    const float h1 = 0.5f * (half ? xv.w : xv.z);   // wire = half + 2
    v2f a1 = { __cosf(h0), __sinf(h0) };
    v2f a2 = { __cosf(h1), __sinf(h1) };

    // D[p, 4r+w] = cos(0.5*(x[p,w] - refs[r,w])), all 16 points x 16 (r,w)
    v8f acc = {};
    acc = __builtin_amdgcn_wmma_f32_16x16x4_f32(false, a1, false, b1,
                                                (short)0, acc, false, false);
    acc = __builtin_amdgcn_wmma_f32_16x16x4_f32(false, a2, false, b2,
                                                (short)0, acc, false, false);

    // Product over the 4 wires: reduce across lanes {4r..4r+3}.
#pragma unroll
    for (int m = 0; m < 8; ++m) {
      float d = acc[m];
      d *= __shfl_xor(d, 1, 32);
      d *= __shfl_xor(d, 2, 32);
      acc[m] = d;
    }

    // Stage (point, r) results into LDS; one lane per group writes.
    // Lane (r, half), vgpr m holds point (m + 8*half). Bank-conflict free.
    if (w == 0) {
#pragma unroll
      for (int m = 0; m < 8; ++m)
        stage[wv][(m + 8 * half) * 4 + r] = acc[m];
    }
    __builtin_amdgcn_wave_barrier();   // wave-synchronous LDS handoff

    if (lane < 16) {
      float4 res = reinterpret_cast<const float4*>(stage[wv])[lane];
      res.x = fabsf(res.x); res.y = fabsf(res.y);
      res.z = fabsf(res.z); res.w = fabsf(res.w);
      out4[pbase + lane] = res;        // 16 lanes x 16B = 256B contiguous
    }
    __builtin_amdgcn_wave_barrier();
  }

  // ---- scalar tail (not hit for B*S = 2^21, kept for safety) --------------
  for (int p = numTiles * 16 + blockIdx.x * kBlock + threadIdx.x;
       p < totalPoints; p += gridDim.x * kBlock) {
    const float4 xv = x4[p];
    float o[4];
#pragma unroll
    for (int rr = 0; rr < 4; ++rr) {
      float pr = __cosf(0.5f * (xv.x - refs[rr * 4 + 0]));
      pr      *= __cosf(0.5f * (xv.y - refs[rr * 4 + 1]));
      pr      *= __cosf(0.5f * (xv.z - refs[rr * 4 + 2]));
      pr      *= __cosf(0.5f * (xv.w - refs[rr * 4 + 3]));
      o[rr] = fabsf(pr);
    }
    out4[p] = make_float4(o[0], o[1], o[2], o[3]);
  }
}

extern "C" void kernel_launch(void* const* d_in, const int* in_sizes, int n_in,
                              void* d_out, int out_size, void* d_ws, size_t ws_size,
                              hipStream_t stream) {
  const float* x    = (const float*)d_in[0];   // (B, S, W) f32
  const float* refs = (const float*)d_in[1];   // (R, W)   f32
  float* out = (float*)d_out;                  // (B, S, R) f32

  const int totalPoints = in_sizes[0] / 4;     // B*S  (W = 4)
  const int numTiles    = totalPoints / 16;    // 16 points per wave-tile

  // ~4 tiles per wave: 2^21 points -> 131072 tiles -> 4096 blocks (32K waves),
  // plenty of in-flight waves to saturate 23.3 TB/s streaming.
  int blocks = (numTiles + 4 * kWavesPerBlock - 1) / (4 * kWavesPerBlock);
  if (blocks < 1) blocks = 1;

  qke_wmma_kernel<<<blocks, kBlock, 0, stream>>>(x, refs, out,
                                                 numTiles, totalPoints);
}